// AdaptiveGraphConv_75617194213767
// MI455X (gfx1250) — compile-verified
//
#include <hip/hip_runtime.h>
#include <hip/hip_bf16.h>
#include <math.h>

// ---------------- problem constants ----------------
#define B_     32
#define CIN    64
#define T_     300
#define V_     25
#define COUT   128
#define INTER_ 32
#define TV     (T_ * V_)           // 7500
#define BT     (B_ * T_)           // 9600
#define NSTAT  ((float)(B_ * T_ * V_))  // 240000 per channel
#define EPS_   1e-5f
#define SOFTMAX_SCALE 0.17677669529663687f   // 1/sqrt(32)

#define WAVES_PER_BLK 4
// LDS strides (even -> 8B-aligned ds_load_b64 fragments; odd-ish mod 64 -> no bank conflicts)
#define TP_STRIDE 66   // 32 x 66 : th (cols 0..31) | phT (cols 32..63)
#define S_STRIDE  34   // 32 x 34 : attention logits / softmax matrix
#define G_STRIDE  16   // 32 x 16 : g tile (w-major)
#define O_STRIDE  17   // 32 x 17 : out tile (v-major)
#define TP_FLOATS (32 * TP_STRIDE)                 // 2112
#define S_FLOATS  (32 * S_STRIDE)                  // 1088
#define WAVE_LDS  (TP_FLOATS + S_FLOATS)           // 3200 floats / wave

typedef __attribute__((ext_vector_type(2))) float v2f;
typedef __attribute__((ext_vector_type(8))) float v8f;

// CDNA5 split-counter wait: order wave-private LDS store->load (no workgroup barrier needed)
__device__ __forceinline__ void lds_fence() {
    asm volatile("s_wait_dscnt 0" ::: "memory");
}

__device__ __forceinline__ v8f splat8(float v) {
    v8f r;
#pragma unroll
    for (int i = 0; i < 8; ++i) r[i] = v;
    return r;
}

// D = A(16x4 f32) * B(4x16 f32) + C(16x16 f32)
__device__ __forceinline__ v8f wmma_f32(v2f a, v2f b, v8f c) {
    return __builtin_amdgcn_wmma_f32_16x16x4_f32(
        /*neg_a=*/false, a, /*neg_b=*/false, b,
        /*c_mod=*/(short)0, c, /*reuse_a=*/false, /*reuse_b=*/false);
}

// =====================================================================
// Kernel 0: zero the stats scratch (sum[128] | sumsq[128])
// =====================================================================
__global__ void agc_zero_stats(float* __restrict__ stats) {
    if (threadIdx.x < 2 * COUT) stats[threadIdx.x] = 0.0f;
}

// =====================================================================
// Kernel 1: fused projections + attention + mix; writes RAW out,
// accumulates per-channel sum / sumsq into stats via atomics.
// One wave per (b,t) slice.
// =====================================================================
__global__ __launch_bounds__(128) void agc_main(
    const float* __restrict__ x,
    const float* __restrict__ thw, const float* __restrict__ thb,
    const float* __restrict__ phw, const float* __restrict__ phb,
    const float* __restrict__ gw,  const float* __restrict__ gb,
    float* __restrict__ out, float* __restrict__ stats)
{
    __shared__ __attribute__((aligned(16))) float lds[WAVES_PER_BLK * WAVE_LDS];
    __shared__ float bsum[COUT];
    __shared__ float bsq[COUT];

    const int tid = threadIdx.x;
    if (tid < COUT) { bsum[tid] = 0.0f; bsq[tid] = 0.0f; }
    __syncthreads();

    const int wave = tid >> 5;
    const int lane = tid & 31;
    const int half = lane >> 4;   // 0: lanes 0-15, 1: lanes 16-31
    const int lm   = lane & 15;

    float* TP = lds + wave * WAVE_LDS;     // 32 x TP_STRIDE
    float* Sb = TP + TP_FLOATS;            // 32 x S_STRIDE
    float* Gb = TP;                        // alias (TP dead after phase C)
    float* Ob = TP + 32 * G_STRIDE;        // alias, disjoint from Gb

    const int slice = blockIdx.x * WAVES_PER_BLK + wave;   // in [0, BT)
    const int b = slice / T_;
    const int t = slice - b * T_;

    // ---------------- Phase A: load A-fragments of Xt (32x64, rows>=25 zero) ----
    // A 16x4 tile layout: lanes 0-15 hold (m=lm, k=4ks+{0,1}); lanes 16-31 hold (m=lm, k=4ks+{2,3})
    v2f a[2][16];
    {
        const float* xb = x + (size_t)b * CIN * TV + (size_t)t * V_;
#pragma unroll
        for (int mt = 0; mt < 2; ++mt) {
            const int v = mt * 16 + lm;
            const bool ok = (v < V_);
            const float* xp = xb + v;
#pragma unroll
            for (int ks = 0; ks < 16; ++ks) {
                const int c0 = ks * 4 + half * 2;
                v2f f;
                f.x = ok ? xp[(size_t)c0 * TV] : 0.0f;
                f.y = ok ? xp[(size_t)(c0 + 1) * TV] : 0.0f;
                a[mt][ks] = f;
            }
        }
    }

    // ---------------- Phase B: TP[:,0:64] = Xt * [theta;phi]^T + bias ----------
#pragma unroll 1
    for (int nt = 0; nt < 4; ++nt) {
        const float* wrow;
        float bias;
        if (nt < 2) { wrow = thw + (nt * 16 + lm) * CIN;       bias = thb[nt * 16 + lm]; }
        else        { wrow = phw + ((nt - 2) * 16 + lm) * CIN; bias = phb[(nt - 2) * 16 + lm]; }
        v8f acc0 = splat8(bias);
        v8f acc1 = splat8(bias);
#pragma unroll
        for (int ks = 0; ks < 16; ++ks) {
            const int kb = ks * 4 + half * 2;
            const v2f bf = *(const v2f*)(wrow + kb);   // B[k][n] = W[n][k]
            acc0 = wmma_f32(a[0][ks], bf, acc0);
            acc1 = wmma_f32(a[1][ks], bf, acc1);
        }
        const int col = nt * 16 + lm;
#pragma unroll
        for (int r = 0; r < 8; ++r) {                  // C/D: vgpr r -> m = half*8 + r
            TP[(half * 8 + r) * TP_STRIDE + col]        = acc0[r];
            TP[(16 + half * 8 + r) * TP_STRIDE + col]   = acc1[r];
        }
    }
    lds_fence();

    // ---------------- Phase C: S = th * ph (logits) ----------------------------
#pragma unroll 1
    for (int nt = 0; nt < 2; ++nt) {
        v8f acc0 = splat8(0.0f);
        v8f acc1 = splat8(0.0f);
#pragma unroll
        for (int ks = 0; ks < 8; ++ks) {
            const int kb = ks * 4 + half * 2;
            const v2f a0 = *(const v2f*)(TP + lm * TP_STRIDE + kb);              // th rows 0-15
            const v2f a1 = *(const v2f*)(TP + (16 + lm) * TP_STRIDE + kb);       // th rows 16-31
            const v2f bf = *(const v2f*)(TP + (nt * 16 + lm) * TP_STRIDE + 32 + kb); // B[k][w]=phT[w][k]
            acc0 = wmma_f32(a0, bf, acc0);
            acc1 = wmma_f32(a1, bf, acc1);
        }
        const int col = nt * 16 + lm;
#pragma unroll
        for (int r = 0; r < 8; ++r) {
            Sb[(half * 8 + r) * S_STRIDE + col]       = acc0[r];
            Sb[(16 + half * 8 + r) * S_STRIDE + col]  = acc1[r];
        }
    }
    lds_fence();

    // ---------------- softmax over w (cols 0..24); one lane per row ------------
    {
        float* row = Sb + lane * S_STRIDE;
        float mx = -3.4e38f;
#pragma unroll
        for (int w = 0; w < V_; ++w) mx = fmaxf(mx, row[w]);
        float e[V_];
        float sum = 0.0f;
#pragma unroll
        for (int w = 0; w < V_; ++w) { e[w] = __expf((row[w] - mx) * SOFTMAX_SCALE); sum += e[w]; }
        const float inv = 1.0f / sum;
#pragma unroll
        for (int w = 0; w < V_; ++w) row[w] = e[w] * inv;
#pragma unroll
        for (int w = V_; w < 32; ++w) row[w] = 0.0f;     // exact padding for A*g
    }
    lds_fence();

    // ---------------- Phase D: per 16-channel tile: g = Xt*gw^T+b, out = A*g ----
    float* outp = out + (size_t)b * COUT * TV + (size_t)t * V_;
#pragma unroll 1
    for (int ot = 0; ot < 8; ++ot) {
        // g tile
        const float* wrow = gw + (ot * 16 + lm) * CIN;
        const float bias = gb[ot * 16 + lm];
        v8f g0 = splat8(bias);
        v8f g1 = splat8(bias);
#pragma unroll
        for (int ks = 0; ks < 16; ++ks) {
            const int kb = ks * 4 + half * 2;
            const v2f bf = *(const v2f*)(wrow + kb);
            g0 = wmma_f32(a[0][ks], bf, g0);
            g1 = wmma_f32(a[1][ks], bf, g1);
        }
#pragma unroll
        for (int r = 0; r < 8; ++r) {                    // Gb[w][j]
            Gb[(half * 8 + r) * G_STRIDE + lm]      = g0[r];
            Gb[(16 + half * 8 + r) * G_STRIDE + lm] = g1[r];
        }
        lds_fence();

        // out tile = A_attn(32x32, cols>=25 zero) * g(32x16)
        v8f o0 = splat8(0.0f);
        v8f o1 = splat8(0.0f);
#pragma unroll
        for (int ks = 0; ks < 8; ++ks) {
            const int kb = ks * 4 + half * 2;
            const v2f a0 = *(const v2f*)(Sb + lm * S_STRIDE + kb);
            const v2f a1 = *(const v2f*)(Sb + (16 + lm) * S_STRIDE + kb);
            v2f bf;
            bf.x = Gb[kb * G_STRIDE + lm];               // B[k][j] = g[w=k][j]
            bf.y = Gb[(kb + 1) * G_STRIDE + lm];
            o0 = wmma_f32(a0, bf, o0);
            o1 = wmma_f32(a1, bf, o1);
        }
#pragma unroll
        for (int r = 0; r < 8; ++r) {                    // Ob[v][j]
            Ob[(half * 8 + r) * O_STRIDE + lm]      = o0[r];
            Ob[(16 + half * 8 + r) * O_STRIDE + lm] = o1[r];
        }
        lds_fence();

        // coalesced-ish global store (25-float runs) + block stats
        for (int idx = lane; idx < V_ * 16; idx += 32) {
            const int oc = idx / V_;
            const int v  = idx - oc * V_;
            const float val = Ob[v * O_STRIDE + oc];
            outp[(size_t)(ot * 16 + oc) * TV + v] = val;
            atomicAdd(&bsum[ot * 16 + oc], val);
            atomicAdd(&bsq[ot * 16 + oc], val * val);
        }
        lds_fence();
    }

    __syncthreads();
    if (tid < COUT) {
        atomicAdd(&stats[tid], bsum[tid]);
        atomicAdd(&stats[COUT + tid], bsq[tid]);
    }
}

// =====================================================================
// Kernel 2: fold stats + gamma/beta into per-channel scale/shift
// =====================================================================
__global__ void agc_finalize(float* __restrict__ stats,
                             const float* __restrict__ gamma,
                             const float* __restrict__ beta) {
    const int c = threadIdx.x;
    if (c < COUT) {
        const float invN = 1.0f / NSTAT;
        const float mean = stats[c] * invN;
        const float var  = stats[COUT + c] * invN - mean * mean;
        const float rstd = rsqrtf(var + EPS_);
        const float sc = gamma[c] * rstd;
        const float sh = beta[c] - mean * sc;
        stats[c] = sc;
        stats[COUT + c] = sh;
    }
}

// =====================================================================
// Kernel 3: apply BN in place on d_out (float4 vectorized; TV%4==0 so a
// float4 never straddles a channel boundary)
// =====================================================================
__global__ __launch_bounds__(256) void agc_apply(float* __restrict__ out,
                                                 const float* __restrict__ stats) {
    __shared__ float sc[COUT];
    __shared__ float sh[COUT];
    if (threadIdx.x < COUT) {
        sc[threadIdx.x] = stats[threadIdx.x];
        sh[threadIdx.x] = stats[COUT + threadIdx.x];
    }
    __syncthreads();
    const size_t total = (size_t)B_ * COUT * TV;
    const size_t gid = (size_t)blockIdx.x * blockDim.x + threadIdx.x;
    const size_t i = gid * 4;
    if (i >= total) return;
    const int ch = (int)((i / (size_t)TV) % COUT);
    float4 v = *(float4*)(out + i);
    const float a = sc[ch], b = sh[ch];
    v.x = fmaf(v.x, a, b);
    v.y = fmaf(v.y, a, b);
    v.z = fmaf(v.z, a, b);
    v.w = fmaf(v.w, a, b);
    *(float4*)(out + i) = v;
}

// =====================================================================
extern "C" void kernel_launch(void* const* d_in, const int* in_sizes, int n_in,
                              void* d_out, int out_size, void* d_ws, size_t ws_size,
                              hipStream_t stream) {
    (void)in_sizes; (void)n_in; (void)out_size; (void)ws_size;
    const float* x     = (const float*)d_in[0];
    const float* thw   = (const float*)d_in[1];
    const float* thb   = (const float*)d_in[2];
    const float* phw   = (const float*)d_in[3];
    const float* phb   = (const float*)d_in[4];
    const float* gw    = (const float*)d_in[5];
    const float* gb    = (const float*)d_in[6];
    const float* gamma = (const float*)d_in[7];
    const float* beta  = (const float*)d_in[8];
    float* out   = (float*)d_out;
    float* stats = (float*)d_ws;   // 256 floats: sum[128] | sumsq[128] -> scale|shift

    agc_zero_stats<<<1, 256, 0, stream>>>(stats);
    agc_main<<<BT / WAVES_PER_BLK, WAVES_PER_BLK * 32, 0, stream>>>(
        x, thw, thb, phw, phb, gw, gb, out, stats);
    agc_finalize<<<1, COUT, 0, stream>>>(stats, gamma, beta);
    const size_t total4 = ((size_t)B_ * COUT * TV) / 4;   // 7,680,000
    agc_apply<<<(unsigned)((total4 + 255) / 256), 256, 0, stream>>>(out, stats);
}